// SurfaceSeparationDistanceElmCom_89910845375152
// MI455X (gfx1250) — compile-verified
//
#include <hip/hip_runtime.h>
#include <math.h>

// ---------------------------------------------------------------------------
// SurfaceSeparationDistanceElmCom for MI455X (gfx1250)
//
// d^2 = |q|^2 + |S|^2 + 2 q.S computed as a 16x16x4 FP32 WMMA:
//   A(16x4)  = [2qx, 2qy, 2qz, |q|^2]   (per slab atom, q = wrap(p) - com)
//   B(4x16)  = [Sx ; Sy ; Sz ; 1  ]     (per lattice shift)
//   C(16x16) = |S|^2 broadcast per column
// Soft-min is accumulated online (running-min rescaling) per lane, then
// merged across the block / grid.  sqrt/exp use the native TRANS32 ops
// (v_sqrt_f32 / v_exp_f32) which co-execute with the XDL/WMMA pipe.
// ---------------------------------------------------------------------------

typedef __attribute__((ext_vector_type(2))) float v2f;
typedef __attribute__((ext_vector_type(8))) float v8f;

#define ACOEF       2.0f        // A = 0.5 / (0.5*0.5)
#define NSHIFT      147         // (2*3+1)*(2*3+1)*(2*1+1)
#define NSHIFT_PAD  160
#define SH_TILES    10
#define QBIG        1.0e18f     // sentinel: masked atoms / padded shifts
#define MN_INIT     3.0e38f
#define PARAM_STRIDE 672        // floats per batch: c[3], cel[9], inv[9], pad, S[160*4]
#define S_OFF       24

// fast exp: v_exp_f32 (exp2) with log2e prescale; exp(-inf)=0, no NaN paths
__device__ __forceinline__ float fast_exp(float x) {
    return __expf(x);
}

// merge two online-softmin partial states (m,num,den) -> into (m1,n1,d1)
__device__ __forceinline__ void smin_merge(float& m1, float& n1, float& d1,
                                           float m2, float n2, float d2) {
    float m  = fminf(m1, m2);
    float s1 = fast_exp(-ACOEF * (m1 - m));
    float s2 = fast_exp(-ACOEF * (m2 - m));
    n1 = (n1 + (m1 - m) * d1) * s1 + (n2 + (m2 - m) * d2) * s2;
    d1 = d1 * s1 + d2 * s2;
    m1 = m;
}

// ---------------------------------------------------------------------------
// Kernel 1: per-batch setup (COM of mol atoms, cell inverse, wrapped COM,
//           shift table with |S|^2) -> workspace param block
// ---------------------------------------------------------------------------
__global__ void k_setup(const float* __restrict__ pos,
                        const float* __restrict__ cel,
                        const float* __restrict__ mas,
                        const int*   __restrict__ elm,
                        const unsigned char* __restrict__ pbc,
                        float* __restrict__ params, int N) {
    int b = blockIdx.x;
    int t = threadIdx.x;
    const float* posb = pos + (size_t)b * N * 3;
    const float* masb = mas + (size_t)b * N;
    const int*   elmb = elm + (size_t)b * N;
    const float* celb = cel + b * 9;

    __shared__ float sm[256], sx[256], sy[256], sz[256];
    float am = 0.f, ax = 0.f, ay = 0.f, az = 0.f;
    for (int n = t; n < N; n += 256) {
        float mm = (elmb[n] == 0) ? masb[n] : 0.0f;   // MOL_KEY == 0
        am += mm;
        ax += mm * posb[n * 3 + 0];
        ay += mm * posb[n * 3 + 1];
        az += mm * posb[n * 3 + 2];
    }
    sm[t] = am; sx[t] = ax; sy[t] = ay; sz[t] = az;
    __syncthreads();
    for (int off = 128; off > 0; off >>= 1) {
        if (t < off) { sm[t] += sm[t+off]; sx[t] += sx[t+off];
                       sy[t] += sy[t+off]; sz[t] += sz[t+off]; }
        __syncthreads();
    }

    float* P = params + (size_t)b * PARAM_STRIDE;
    if (t == 0) {
        float M[9];
        #pragma unroll
        for (int i = 0; i < 9; ++i) M[i] = celb[i];
        // 3x3 inverse via adjugate
        float c00 =  M[4]*M[8] - M[5]*M[7];
        float c01 = -(M[3]*M[8] - M[5]*M[6]);
        float c02 =  M[3]*M[7] - M[4]*M[6];
        float det = M[0]*c00 + M[1]*c01 + M[2]*c02;
        float inv[9];
        inv[0] = c00 / det;
        inv[1] = (M[2]*M[7] - M[1]*M[8]) / det;
        inv[2] = (M[1]*M[5] - M[2]*M[4]) / det;
        inv[3] = c01 / det;
        inv[4] = (M[0]*M[8] - M[2]*M[6]) / det;
        inv[5] = (M[2]*M[3] - M[0]*M[5]) / det;
        inv[6] = c02 / det;
        inv[7] = (M[1]*M[6] - M[0]*M[7]) / det;
        inv[8] = (M[0]*M[4] - M[1]*M[3]) / det;

        float msum = sm[0];
        float com[3] = { sx[0]/msum, sy[0]/msum, sz[0]/msum };
        // wrap COM
        float spc[3];
        #pragma unroll
        for (int e = 0; e < 3; ++e) {
            float f = com[0]*inv[0*3+e] + com[1]*inv[1*3+e] + com[2]*inv[2*3+e];
            spc[e] = (pbc[b*3+e] != 0) ? floorf(f) : 0.0f;
        }
        #pragma unroll
        for (int d = 0; d < 3; ++d)
            P[d] = com[d] - (spc[0]*M[0*3+d] + spc[1]*M[1*3+d] + spc[2]*M[2*3+d]);
        #pragma unroll
        for (int i = 0; i < 9; ++i) { P[3 + i] = M[i]; P[12 + i] = inv[i]; }
    }

    // shift table: s = ((i+3)*7 + (j+3))*3 + (k+1)
    if (t < NSHIFT_PAD) {
        float Sx, Sy, Sz, S2;
        if (t < NSHIFT) {
            float fi = (float)(t / 21 - 3);
            float fj = (float)((t / 3) % 7 - 3);
            float fk = (float)(t % 3 - 1);
            Sx = fi*celb[0] + fj*celb[3] + fk*celb[6];
            Sy = fi*celb[1] + fj*celb[4] + fk*celb[7];
            Sz = fi*celb[2] + fj*celb[5] + fk*celb[8];
            S2 = Sx*Sx + Sy*Sy + Sz*Sz;
        } else {                       // padding -> huge distance, zero weight
            Sx = 0.f; Sy = 0.f; Sz = QBIG; S2 = QBIG * QBIG;
        }
        P[S_OFF + t*4 + 0] = Sx;
        P[S_OFF + t*4 + 1] = Sy;
        P[S_OFF + t*4 + 2] = Sz;
        P[S_OFF + t*4 + 3] = S2;
    }
}

// ---------------------------------------------------------------------------
// Kernel 2: distances via WMMA f32 16x16x4 + online softmin, one partial
//           (min,num,den) per block.  grid = (chunks, B), block = 256 (8 waves)
// ---------------------------------------------------------------------------
__global__ void k_main(const float* __restrict__ pos,
                       const int*   __restrict__ elm,
                       const unsigned char* __restrict__ pbc,
                       const float* __restrict__ params,
                       float* __restrict__ partials, int N, int chunks) {
    int b     = blockIdx.y;
    int chunk = blockIdx.x;
    int t     = threadIdx.x;
    int lane  = t & 31;          // wave32
    int wave  = t >> 5;
    const float* P = params + (size_t)b * PARAM_STRIDE;

    __shared__ float sS[NSHIFT_PAD * 4];
    __shared__ float sCel[9], sInv[9], sC[3];
    for (int i = t; i < NSHIFT_PAD * 4; i += 256) sS[i] = P[S_OFF + i];
    if (t < 9) { sCel[t] = P[3 + t]; sInv[t] = P[12 + t]; }
    if (t < 3) { sC[t] = P[t]; }
    __syncthreads();

    const float* posb = pos + (size_t)b * N * 3;
    const int*   elmb = elm + (size_t)b * N;
    bool pb0 = pbc[b*3+0] != 0, pb1 = pbc[b*3+1] != 0, pb2 = pbc[b*3+2] != 0;

    int  m_local = lane & 15;
    bool hi      = lane >= 16;
    int  atomBase = chunk * 256;

    float mn = MN_INIT, num = 0.0f, den = 0.0f;

    for (int tt = 0; tt < 2; ++tt) {                 // 2 atom tiles per wave
        int atom  = atomBase + (wave * 2 + tt) * 16 + m_local;
        bool valid = atom < N;
        int  ai   = valid ? atom : (N - 1);
        float px = posb[ai*3+0], py = posb[ai*3+1], pz = posb[ai*3+2];
        bool slab = valid && (elmb[ai] == 1);        // SLB_KEY == 1

        // wrap(p)
        float f0 = px*sInv[0] + py*sInv[3] + pz*sInv[6];
        float f1 = px*sInv[1] + py*sInv[4] + pz*sInv[7];
        float f2 = px*sInv[2] + py*sInv[5] + pz*sInv[8];
        float sp0 = pb0 ? floorf(f0) : 0.0f;
        float sp1 = pb1 ? floorf(f1) : 0.0f;
        float sp2 = pb2 ? floorf(f2) : 0.0f;
        float wx = px - (sp0*sCel[0] + sp1*sCel[3] + sp2*sCel[6]);
        float wy = py - (sp0*sCel[1] + sp1*sCel[4] + sp2*sCel[7]);
        float wz = pz - (sp0*sCel[2] + sp1*sCel[5] + sp2*sCel[8]);

        float qx = slab ? (wx - sC[0]) : 0.0f;
        float qy = slab ? (wy - sC[1]) : 0.0f;
        float qz = slab ? (wz - sC[2]) : QBIG;       // masked -> huge distance
        float q2 = qx*qx + qy*qy + qz*qz;

        // A(16x4) f32 layout: lanes 0-15 hold K0,K1; lanes 16-31 hold K2,K3
        v2f a;
        a.x = hi ? (2.0f * qz) : (2.0f * qx);
        a.y = hi ? q2          : (2.0f * qy);

        for (int st = 0; st < SH_TILES; ++st) {
            int sIdx = st * 16 + m_local;            // column n = lane&15
            float Sx = sS[sIdx*4+0], Sy = sS[sIdx*4+1];
            float Sz = sS[sIdx*4+2], S2 = sS[sIdx*4+3];
            v2f bv;                                   // B(4x16), K mirrored
            bv.x = hi ? Sz   : Sx;
            bv.y = hi ? 1.0f : Sy;
            v8f cm = { S2, S2, S2, S2, S2, S2, S2, S2 };  // |S|^2 per column

            v8f d2 = __builtin_amdgcn_wmma_f32_16x16x4_f32(
                false, a, false, bv, (short)0, cm, false, false);

            #pragma unroll
            for (int r = 0; r < 8; ++r) {
                // native TRANS32 sqrt (v_sqrt_f32), ~1 ULP
                float dist = __builtin_amdgcn_sqrtf(fmaxf(d2[r], 0.0f));
                float nm = fminf(mn, dist);
                float s1 = fast_exp(-ACOEF * (mn - nm)); // rescale old state
                num = (num + (mn - nm) * den) * s1;
                den = den * s1;
                float w = fast_exp(-ACOEF * (dist - nm));
                num += (dist - nm) * w;
                den += w;
                mn = nm;
            }
        }
    }

    // block reduce 256 partial softmin states
    __shared__ float rm[256], rn[256], rd[256];
    rm[t] = mn; rn[t] = num; rd[t] = den;
    __syncthreads();
    for (int off = 128; off > 0; off >>= 1) {
        if (t < off) {
            float m1 = rm[t], n1 = rn[t], d1 = rd[t];
            smin_merge(m1, n1, d1, rm[t+off], rn[t+off], rd[t+off]);
            rm[t] = m1; rn[t] = n1; rd[t] = d1;
        }
        __syncthreads();
    }
    if (t == 0) {
        float* pp = partials + ((size_t)b * chunks + chunk) * 3;
        pp[0] = rm[0]; pp[1] = rn[0]; pp[2] = rd[0];
    }
}

// ---------------------------------------------------------------------------
// Kernel 3: merge chunk partials per batch -> out[b]
// ---------------------------------------------------------------------------
__global__ void k_final(const float* __restrict__ partials,
                        float* __restrict__ out, int chunks) {
    int b = blockIdx.x;
    if (threadIdx.x != 0) return;
    float mn = MN_INIT, num = 0.0f, den = 0.0f;
    for (int c = 0; c < chunks; ++c) {
        const float* pp = partials + ((size_t)b * chunks + c) * 3;
        smin_merge(mn, num, den, pp[0], pp[1], pp[2]);
    }
    out[b] = (den > 0.0f) ? (num / den + mn) : mn;
}

// ---------------------------------------------------------------------------
extern "C" void kernel_launch(void* const* d_in, const int* in_sizes, int n_in,
                              void* d_out, int out_size, void* d_ws, size_t ws_size,
                              hipStream_t stream) {
    const float* pos = (const float*)d_in[0];          // (B,N,3) f32
    const float* cel = (const float*)d_in[1];          // (B,3,3) f32
    const float* mas = (const float*)d_in[2];          // (B,N)   f32
    const int*   elm = (const int*)d_in[3];            // (B,N)   i32
    const unsigned char* pbc = (const unsigned char*)d_in[4]; // (B,3) bool8
    float* out = (float*)d_out;                        // (B,1)   f32
    float* ws  = (float*)d_ws;

    int B = in_sizes[1] / 9;
    int N = in_sizes[2] / B;
    int chunks = (N + 255) / 256;

    float* params   = ws;
    float* partials = ws + (size_t)B * PARAM_STRIDE;

    k_setup<<<B, 256, 0, stream>>>(pos, cel, mas, elm, pbc, params, N);
    k_main<<<dim3(chunks, B), 256, 0, stream>>>(pos, elm, pbc, params,
                                                partials, N, chunks);
    k_final<<<B, 32, 0, stream>>>(partials, out, chunks);
}